// Prototypical_72679436583764
// MI455X (gfx1250) — compile-verified
//
#include <hip/hip_runtime.h>
#include <hip/hip_bf16.h>
#include <math.h>

// ---------------------------------------------------------------------------
// Prototypical episodic network on gfx1250 (MI455X), f32 WMMA path.
//   B=16, N=512, D=256, H=8, DK=32, C=16
// All GEMMs use V_WMMA_F32_16X16X4_F32 (one 16x16 D-tile per wave32).
// Final stage uses the symmetric-S identity:
//   out[n,c] = (x_n . P_c - |x_n|^2 * [lab_n==c]) / (counts_c - [lab_n==c])
// Total ~22 GFLOP, ~80 MB activations (L2-resident on 192 MB L2).
// ---------------------------------------------------------------------------

typedef __attribute__((ext_vector_type(2))) float v2f;
typedef __attribute__((ext_vector_type(8))) float v8f;

#define B_ 16
#define N_ 512
#define D_ 256
#define H_ 8
#define DK_ 32
#define C_ 16

__device__ __forceinline__ v8f wmma4(v2f a, v2f b, v8f c) {
  // (neg_a, A, neg_b, B, c_mod, C, reuse_a, reuse_b)
  return __builtin_amdgcn_wmma_f32_16x16x4_f32(false, a, false, b, (short)0, c,
                                               false, false);
}

__device__ __forceinline__ float gelu_f(float x) {
  return 0.5f * x * (1.0f + erff(x * 0.70710678118654752f));
}

// ---------------------------------------------------------------------------
// Generic batched GEMM: C[b] = act(A[b] @ W + bias)
// A: [M,K] row-major (lda), W: [K,N] row-major (ldb), C: [M,N] (ldc)
// Block = 256 threads (8 waves) -> 32x64 output tile (2x4 waves of 16x16).
// grid = (N/64, M/32, batch); strideA/strideC are PER-EPISODE element strides.
// ---------------------------------------------------------------------------
template <int ACT, bool BIAS>
__global__ void gemm_nn_wmma(const float* __restrict__ A,
                             const float* __restrict__ W,
                             const float* __restrict__ bias,
                             float* __restrict__ C, int K, int lda, int ldb,
                             int ldc, long strideA, long strideC) {
  A += blockIdx.z * strideA;
  C += blockIdx.z * strideC;
  const int lane = threadIdx.x & 31;
  const int wave = threadIdx.x >> 5;
  const int hb = lane >> 4;   // K-half selector (0: K=0,1 ; 1: K=2,3)
  const int r = lane & 15;    // row (A) / col (B,C) within tile
  const int m0 = blockIdx.y * 32 + (wave >> 2) * 16;
  const int n0 = blockIdx.x * 64 + (wave & 3) * 16;

  v8f acc = {};
#pragma unroll 4
  for (int kk = 0; kk < K; kk += 4) {
    const int ka = kk + 2 * hb;
    v2f a;
    a.x = A[(m0 + r) * lda + ka];
    a.y = A[(m0 + r) * lda + ka + 1];
    v2f b;
    b.x = W[ka * ldb + n0 + r];
    b.y = W[(ka + 1) * ldb + n0 + r];
    acc = wmma4(a, b, acc);
  }
#pragma unroll
  for (int j = 0; j < 8; ++j) {
    const int row = m0 + j + 8 * hb;
    const int col = n0 + r;
    float v = acc[j];
    if (BIAS) v += bias[col];
    if (ACT == 1) v = gelu_f(v);
    C[row * ldc + col] = v;
  }
}

// ---------------------------------------------------------------------------
// Fused attention: one wave per (b, h, 16-row strip).
// scores strip [16 x 512] staged in LDS (stride 520 -> conflict-free),
// softmax via wave32 shuffles, attn@V back through WMMA.
// grid = (N/16, H, B), block = 32
// ---------------------------------------------------------------------------
#define SROW 520
__global__ void attention_wmma(const float* __restrict__ q,
                               const float* __restrict__ k,
                               const float* __restrict__ v,
                               float* __restrict__ out) {
  __shared__ float s[16 * SROW];
  const int lane = threadIdx.x;
  const int hb = lane >> 4;
  const int r = lane & 15;
  const int m0 = blockIdx.x * 16;
  const int h = blockIdx.y;
  const int b = blockIdx.z;
  const float* qb = q + (long)b * N_ * D_ + h * DK_;
  const float* kb = k + (long)b * N_ * D_ + h * DK_;
  const float* vb = v + (long)b * N_ * D_ + h * DK_;
  const float scale = 0.17677669529663687f;  // 1/sqrt(DK)

  // ---- scores = (q_strip @ K^T) * scale -> LDS ----
  for (int nt = 0; nt < N_ / 16; ++nt) {
    const int n0 = nt * 16;
    v8f acc = {};
#pragma unroll
    for (int kk = 0; kk < DK_; kk += 4) {
      const int ka = kk + 2 * hb;
      v2f a;
      a.x = qb[(m0 + r) * D_ + ka];
      a.y = qb[(m0 + r) * D_ + ka + 1];
      v2f bf;  // B[d][n] = K[n][d]
      bf.x = kb[(n0 + r) * D_ + ka];
      bf.y = kb[(n0 + r) * D_ + ka + 1];
      acc = wmma4(a, bf, acc);
    }
#pragma unroll
    for (int j = 0; j < 8; ++j)
      s[(j + 8 * hb) * SROW + n0 + r] = acc[j] * scale;
  }
  __syncthreads();

  // ---- row-wise softmax over 512 cols (wave32 shuffle reductions) ----
  for (int row = 0; row < 16; ++row) {
    float m = -INFINITY;
    for (int c = lane; c < N_; c += 32) m = fmaxf(m, s[row * SROW + c]);
#pragma unroll
    for (int off = 16; off > 0; off >>= 1) m = fmaxf(m, __shfl_xor(m, off, 32));
    float sum = 0.0f;
    for (int c = lane; c < N_; c += 32) {
      float e = expf(s[row * SROW + c] - m);
      s[row * SROW + c] = e;
      sum += e;
    }
#pragma unroll
    for (int off = 16; off > 0; off >>= 1) sum += __shfl_xor(sum, off, 32);
    const float inv = 1.0f / sum;
    for (int c = lane; c < N_; c += 32) s[row * SROW + c] *= inv;
  }
  __syncthreads();

  // ---- out_strip = attn @ V (A fragments from LDS) ----
#pragma unroll
  for (int t = 0; t < 2; ++t) {  // two 16-wide column tiles of DK=32
    v8f acc = {};
    for (int kk = 0; kk < N_; kk += 4) {
      const int ka = kk + 2 * hb;
      v2f a;
      a.x = s[r * SROW + ka];
      a.y = s[r * SROW + ka + 1];
      v2f bf;
      bf.x = vb[ka * D_ + t * 16 + r];
      bf.y = vb[(ka + 1) * D_ + t * 16 + r];
      acc = wmma4(a, bf, acc);
    }
#pragma unroll
    for (int j = 0; j < 8; ++j)
      out[((long)b * N_ + m0 + j + 8 * hb) * D_ + h * DK_ + t * 16 + r] =
          acc[j];
  }
}

// ---------------------------------------------------------------------------
// aggr[b,d] = gelu(mean_n x3[b,n,d]) ; grid = B, block = D
// ---------------------------------------------------------------------------
__global__ void mean_gelu_kernel(const float* __restrict__ x3,
                                 float* __restrict__ aggr) {
  const int b = blockIdx.x, d = threadIdx.x;
  float acc = 0.0f;
  for (int n = 0; n < N_; ++n) acc += x3[((long)b * N_ + n) * D_ + d];
  aggr[b * D_ + d] = gelu_f(acc * (1.0f / N_));
}

// r[b,d2] = aggr[b,:] @ res_W[:,d2] + res_b[d2] ; grid = B, block = D
__global__ void res_linear_kernel(const float* __restrict__ aggr,
                                  const float* __restrict__ resW,
                                  const float* __restrict__ resb,
                                  float* __restrict__ rvec) {
  const int b = blockIdx.x, d2 = threadIdx.x;
  float acc = resb[d2];
  for (int d = 0; d < D_; ++d) acc += aggr[b * D_ + d] * resW[d * D_ + d2];
  rvec[b * D_ + d2] = acc;
}

// e2[b,n,j] = gelu(j < D ? r[b,j] : e[b,n,j-D]) ; flat elementwise
__global__ void build_e2_kernel(const float* __restrict__ e,
                                const float* __restrict__ rvec,
                                float* __restrict__ e2) {
  const long idx = (long)blockIdx.x * blockDim.x + threadIdx.x;
  const int j = (int)(idx & (2 * D_ - 1));
  const long bn = idx / (2 * D_);
  const int n = (int)(bn & (N_ - 1));
  const int b = (int)(bn / N_);
  const float val =
      (j < D_) ? rvec[b * D_ + j] : e[((long)b * N_ + n) * D_ + (j - D_)];
  e2[idx] = gelu_f(val);
}

// ---------------------------------------------------------------------------
// Class prototype sums via WMMA: P[b,c,d] = sum_n [lab==c] * xff[b,n,d]
// A-fragment (one-hot^T) synthesized from labels. grid = (D/16, B), block=32
// ---------------------------------------------------------------------------
__global__ void proto_wmma(const float* __restrict__ xff,
                           const int* __restrict__ labels,
                           float* __restrict__ P) {
  const int lane = threadIdx.x;
  const int hb = lane >> 4;
  const int r = lane & 15;
  const int n0 = blockIdx.x * 16;
  const int b = blockIdx.y;
  const int* lb = labels + b * N_;
  const float* xb = xff + (long)b * N_ * D_;
  v8f acc = {};
  for (int kk = 0; kk < N_; kk += 4) {
    const int ka = kk + 2 * hb;
    v2f a;  // A[c][n] = (lab[n]==c), row c = r
    a.x = (lb[ka] == r) ? 1.0f : 0.0f;
    a.y = (lb[ka + 1] == r) ? 1.0f : 0.0f;
    v2f bf;
    bf.x = xb[(long)ka * D_ + n0 + r];
    bf.y = xb[(long)(ka + 1) * D_ + n0 + r];
    acc = wmma4(a, bf, acc);
  }
#pragma unroll
  for (int j = 0; j < 8; ++j)
    P[(long)b * C_ * D_ + (j + 8 * hb) * D_ + n0 + r] = acc[j];
}

// counts[b,c] = #n with lab==c ; grid = B, block = 32
__global__ void counts_kernel(const int* __restrict__ labels,
                              float* __restrict__ counts) {
  const int b = blockIdx.x, c = threadIdx.x;
  if (c < C_) {
    int cnt = 0;
    for (int n = 0; n < N_; ++n) cnt += (labels[b * N_ + n] == c);
    counts[b * C_ + c] = (float)cnt;
  }
}

// sq[b,n] = |xff[b,n,:]|^2 ; one wave per row, 8 waves/block
__global__ void sq_kernel(const float* __restrict__ xff,
                          float* __restrict__ sq) {
  const int wid = blockIdx.x * 8 + (threadIdx.x >> 5);
  const int lane = threadIdx.x & 31;
  const int b = wid / N_, n = wid % N_;
  const float* xrow = xff + ((long)b * N_ + n) * D_;
  float acc = 0.0f;
  for (int d = lane; d < D_; d += 32) {
    const float x = xrow[d];
    acc += x * x;
  }
#pragma unroll
  for (int off = 16; off > 0; off >>= 1) acc += __shfl_xor(acc, off, 32);
  if (lane == 0) sq[b * N_ + n] = acc;
}

// ---------------------------------------------------------------------------
// Final: G = xff @ P^T via WMMA, fused leave-one-out epilogue.
// grid = (N/16, B), block = 32. N-tile == C == 16 so one column tile.
// ---------------------------------------------------------------------------
__global__ void final_wmma(const float* __restrict__ xff,
                           const float* __restrict__ P,
                           const float* __restrict__ sq,
                           const float* __restrict__ counts,
                           const int* __restrict__ labels,
                           float* __restrict__ out) {
  const int lane = threadIdx.x;
  const int hb = lane >> 4;
  const int r = lane & 15;
  const int m0 = blockIdx.x * 16;
  const int b = blockIdx.y;
  const float* xb = xff + (long)b * N_ * D_;
  const float* Pb = P + (long)b * C_ * D_;
  v8f acc = {};
#pragma unroll 4
  for (int kk = 0; kk < D_; kk += 4) {
    const int ka = kk + 2 * hb;
    v2f a;
    a.x = xb[(m0 + r) * D_ + ka];
    a.y = xb[(m0 + r) * D_ + ka + 1];
    v2f bf;  // B[d][c] = P[c][d], col c = r
    bf.x = Pb[r * D_ + ka];
    bf.y = Pb[r * D_ + ka + 1];
    acc = wmma4(a, bf, acc);
  }
#pragma unroll
  for (int j = 0; j < 8; ++j) {
    const int row = m0 + j + 8 * hb;
    const int c = r;
    const int lab = labels[b * N_ + row];
    const float self = (lab == c) ? sq[b * N_ + row] : 0.0f;
    const float denom = counts[b * C_ + c] - ((lab == c) ? 1.0f : 0.0f);
    out[((long)b * N_ + row) * C_ + c] = (acc[j] - self) / denom;
  }
}

// ---------------------------------------------------------------------------
extern "C" void kernel_launch(void* const* d_in, const int* in_sizes, int n_in,
                              void* d_out, int out_size, void* d_ws,
                              size_t ws_size, hipStream_t stream) {
  const float* e = (const float*)d_in[0];
  const int* labels = (const int*)d_in[1];
  const float* Wq0 = (const float*)d_in[2];
  const float* Wk0 = (const float*)d_in[3];
  const float* Wv0 = (const float*)d_in[4];
  const float* Wo0 = (const float*)d_in[5];
  const float* Wq1 = (const float*)d_in[6];
  const float* Wk1 = (const float*)d_in[7];
  const float* Wv1 = (const float*)d_in[8];
  const float* Wo1 = (const float*)d_in[9];
  const float* res_W = (const float*)d_in[10];
  const float* res_b = (const float*)d_in[11];
  const float* ff_W1 = (const float*)d_in[12];
  const float* ff_b1 = (const float*)d_in[13];
  const float* ff_W2 = (const float*)d_in[14];
  const float* ff_b2 = (const float*)d_in[15];
  float* out = (float*)d_out;

  // workspace carve-up (~80 MB total; all activations L2-resident)
  const long per = (long)N_ * D_;       // per-episode activation stride (131072)
  const long act = (long)B_ * per;      // full buffer size (2,097,152 floats)
  float* w = (float*)d_ws;
  float* qbuf = w; w += act;
  float* kbuf = w; w += act;
  float* vbuf = w; w += act;
  float* attnb = w; w += act;
  float* x1 = w; w += act;            // post Wo0 + gelu
  float* x3 = w; w += act;            // post Wo1
  float* e2 = w; w += 2 * act;        // [B,N,2D]
  float* h1 = w; w += act;
  float* xff = w; w += act;
  float* aggr = w; w += (long)B_ * D_;
  float* rvec = w; w += (long)B_ * D_;
  float* P = w; w += (long)B_ * C_ * D_;
  float* counts = w; w += (long)B_ * C_;
  float* sq = w; w += (long)B_ * N_;

  const dim3 gG(D_ / 64, N_ / 32, B_);   // GEMM grid for [512 x 256] outputs
  const dim3 gA(N_ / 16, H_, B_);        // attention grid

  // ---- MHA layer 0 ----
  gemm_nn_wmma<0, false><<<gG, 256, 0, stream>>>(e, Wq0, nullptr, qbuf, D_, D_, D_, D_, per, per);
  gemm_nn_wmma<0, false><<<gG, 256, 0, stream>>>(e, Wk0, nullptr, kbuf, D_, D_, D_, D_, per, per);
  gemm_nn_wmma<0, false><<<gG, 256, 0, stream>>>(e, Wv0, nullptr, vbuf, D_, D_, D_, D_, per, per);
  attention_wmma<<<gA, 32, 0, stream>>>(qbuf, kbuf, vbuf, attnb);
  gemm_nn_wmma<1, false><<<gG, 256, 0, stream>>>(attnb, Wo0, nullptr, x1, D_, D_, D_, D_, per, per);  // fused gelu

  // ---- MHA layer 1 ----
  gemm_nn_wmma<0, false><<<gG, 256, 0, stream>>>(x1, Wq1, nullptr, qbuf, D_, D_, D_, D_, per, per);
  gemm_nn_wmma<0, false><<<gG, 256, 0, stream>>>(x1, Wk1, nullptr, kbuf, D_, D_, D_, D_, per, per);
  gemm_nn_wmma<0, false><<<gG, 256, 0, stream>>>(x1, Wv1, nullptr, vbuf, D_, D_, D_, D_, per, per);
  attention_wmma<<<gA, 32, 0, stream>>>(qbuf, kbuf, vbuf, attnb);
  gemm_nn_wmma<0, false><<<gG, 256, 0, stream>>>(attnb, Wo1, nullptr, x3, D_, D_, D_, D_, per, per);

  // ---- set summary + residual linear + e2 ----
  mean_gelu_kernel<<<B_, D_, 0, stream>>>(x3, aggr);
  res_linear_kernel<<<B_, D_, 0, stream>>>(aggr, res_W, res_b, rvec);
  build_e2_kernel<<<(B_ * N_ * 2 * D_) / 256, 256, 0, stream>>>(e, rvec, e2);

  // ---- feed-forward (e2 per-episode stride = N*2D) ----
  gemm_nn_wmma<1, true><<<gG, 256, 0, stream>>>(e2, ff_W1, ff_b1, h1, 2 * D_, 2 * D_, D_, D_, 2 * per, per);
  gemm_nn_wmma<0, true><<<gG, 256, 0, stream>>>(h1, ff_W2, ff_b2, xff, D_, D_, D_, D_, per, per);

  // ---- leave-one-out class means (symmetric-S identity) ----
  proto_wmma<<<dim3(D_ / 16, B_), 32, 0, stream>>>(xff, labels, P);
  counts_kernel<<<B_, 32, 0, stream>>>(labels, counts);
  sq_kernel<<<(B_ * N_) / 8, 256, 0, stream>>>(xff, sq);
  final_wmma<<<dim3(N_ / 16, B_), 32, 0, stream>>>(xff, P, sq, counts, labels, out);
}